// CycleEmbedding0_14267881357891
// MI455X (gfx1250) — compile-verified
//
#include <hip/hip_runtime.h>

typedef __attribute__((ext_vector_type(2))) float v2f;
typedef __attribute__((ext_vector_type(8))) float v8f;

#define HIDDEN   128
#define NTYPES   28
#define TSTRIDE  32   // count row stride, padded 28 -> 32 (pad entries stay 0)

// ---------------------------------------------------------------------------
// Phase 1: count[c, t] = #incidences with cycle c and atom-type t
// Only 4 bytes of atomic traffic per incidence (vs 512 B in naive scatter);
// the 64 MB count matrix is L2-resident (192 MB L2).
// ---------------------------------------------------------------------------
__global__ void cyc_count_kernel(const int* __restrict__ x,
                                 const int* __restrict__ a2c,
                                 int* __restrict__ counts,
                                 int E) {
    int idx    = blockIdx.x * blockDim.x + threadIdx.x;
    int stride = gridDim.x * blockDim.x;
    for (int e = idx; e < E; e += stride) {
        int atom = a2c[e];          // row 0 of atom_to_cycle
        int cyc  = a2c[E + e];      // row 1
        int t    = x[atom];         // 0..27
        atomicAdd(&counts[(long long)cyc * TSTRIDE + t], 1);
    }
}

// ---------------------------------------------------------------------------
// Phase 2: out = counts(500k x 28) @ W(28 x 128) via V_WMMA_F32_16X16X4_F32.
// One wave handles a 16-cycle x 128-hidden strip: 7 K-steps x 8 N-tiles = 56
// WMMAs. A layout (32-bit 16x4): lanes 0-15 -> K=0,1 in v0,v1; lanes 16-31
// -> K=2,3. C/D layout: VGPR r holds M = r + 8*(lane>=16), N = lane&15.
// Full tiles take a wave-uniform unguarded store path (no per-store EXEC
// juggling); only a partial tail tile takes the guarded path.
// ---------------------------------------------------------------------------
__global__ void cyc_gemm_wmma_kernel(const int* __restrict__ counts,
                                     const float* __restrict__ W,
                                     float* __restrict__ out,
                                     int numTiles, int numCycles) {
    const int lane    = threadIdx.x & 31;
    const int waveId  = threadIdx.x >> 5;
    const int tile    = blockIdx.x * (blockDim.x >> 5) + waveId;
    if (tile >= numTiles) return;          // wave-uniform: EXEC stays all-1s

    const int halfSel = lane >> 4;         // 0 for lanes 0-15, 1 for 16-31
    const int sub     = lane & 15;
    const long long rowBase = (long long)tile * 16;
    const bool fullTile = (rowBase + 16) <= (long long)numCycles; // wave-uniform

    // A fragments for all 7 K-steps (shared by every N-tile) — load once.
    v2f afrag[7];
    const int* arow = counts + (rowBase + sub) * TSTRIDE + halfSel * 2;
#pragma unroll
    for (int k = 0; k < 7; ++k) {
        afrag[k].x = (float)arow[k * 4];
        afrag[k].y = (float)arow[k * 4 + 1];
    }

    // Per-lane store base: row (halfSel*8), column sub of this strip.
    float* obase = out + (rowBase + halfSel * 8) * HIDDEN + sub;

#pragma unroll
    for (int nt = 0; nt < 8; ++nt) {
        const int colN = nt * 16 + sub;    // output column this lane produces
        v8f acc = {};
#pragma unroll
        for (int k = 0; k < 7; ++k) {
            // B fragment: W[k*4 + halfSel*2 + {0,1}][colN], row stride 128
            const float* bp = W + (k * 4 + halfSel * 2) * HIDDEN + colN;
            v2f b;
            b.x = bp[0];
            b.y = bp[HIDDEN];
            acc = __builtin_amdgcn_wmma_f32_16x16x4_f32(
                /*neg_a=*/false, afrag[k],
                /*neg_b=*/false, b,
                /*c_mod=*/(short)0, acc,
                /*reuse_a=*/false, /*reuse_b=*/false);
        }
        if (fullTile) {
            // Unguarded: 8 stores, immediate offsets (row stride 512 B).
            float* op = obase + nt * 16;
#pragma unroll
            for (int r = 0; r < 8; ++r)
                op[r * HIDDEN] = acc[r];
        } else {
            // Tail tile only: per-row bounds check.
#pragma unroll
            for (int r = 0; r < 8; ++r) {
                long long row = rowBase + halfSel * 8 + r;
                if (row < numCycles)
                    out[row * HIDDEN + nt * 16 + sub] = acc[r];
            }
        }
    }
}

// ---------------------------------------------------------------------------
// Fallback (workspace too small): classic atomic scatter with W in LDS.
// ---------------------------------------------------------------------------
__global__ void cyc_scatter_kernel(const int* __restrict__ x,
                                   const int* __restrict__ a2c,
                                   const float* __restrict__ W,
                                   float* __restrict__ out,
                                   int E) {
    __shared__ float sW[NTYPES * HIDDEN];  // 14 KB of the 320 KB WGP LDS
    for (int i = threadIdx.x; i < NTYPES * HIDDEN; i += blockDim.x)
        sW[i] = W[i];
    __syncthreads();

    const int lane   = threadIdx.x & 31;
    const int warp   = (blockIdx.x * blockDim.x + threadIdx.x) >> 5;
    const int nwarps = (gridDim.x * blockDim.x) >> 5;
    for (int e = warp; e < E; e += nwarps) {
        int atom = a2c[e];
        int cyc  = a2c[E + e];
        int t    = x[atom];
        const float* wrow = sW + t * HIDDEN;
        float* orow = out + (long long)cyc * HIDDEN;
#pragma unroll
        for (int h = lane; h < HIDDEN; h += 32)
            atomicAdd(&orow[h], wrow[h]);
    }
}

static inline int imin_host(int a, int b) { return a < b ? a : b; }

extern "C" void kernel_launch(void* const* d_in, const int* in_sizes, int n_in,
                              void* d_out, int out_size, void* d_ws, size_t ws_size,
                              hipStream_t stream) {
    const int*   x   = (const int*)d_in[0];      // [N_ATOMS]
    const int*   a2c = (const int*)d_in[1];      // [2, E] flat
    const float* W   = (const float*)d_in[2];    // [28, 128]
    float*       out = (float*)d_out;            // [NUM_CYCLES, 128]

    const int E         = in_sizes[1] / 2;
    const int numCycles = out_size / HIDDEN;
    const int numTiles  = (numCycles + 15) / 16;
    const size_t countsBytes = (size_t)numTiles * 16 * TSTRIDE * sizeof(int);

    if (ws_size >= countsBytes) {
        int* counts = (int*)d_ws;
        hipMemsetAsync(counts, 0, countsBytes, stream);

        {
            const int threads = 256;
            const int blocks  = imin_host((E + threads - 1) / threads, 32768);
            cyc_count_kernel<<<blocks, threads, 0, stream>>>(x, a2c, counts, E);
        }
        {
            const int threads       = 256;         // 8 waves per block
            const int wavesPerBlock = threads / 32;
            const int blocks = (numTiles + wavesPerBlock - 1) / wavesPerBlock;
            cyc_gemm_wmma_kernel<<<blocks, threads, 0, stream>>>(
                counts, W, out, numTiles, numCycles);
        }
    } else {
        hipMemsetAsync(out, 0, (size_t)out_size * sizeof(float), stream);
        const int threads     = 256;
        const int incPerBlock = threads / 32;
        const int blocks = imin_host((E + incPerBlock - 1) / incPerBlock, 65535);
        cyc_scatter_kernel<<<blocks, threads, 0, stream>>>(x, a2c, W, out, E);
    }
}